// GraphSAGEModel_44547400794522
// MI455X (gfx1250) — compile-verified
//
#include <hip/hip_runtime.h>

// GraphSAGE (2-layer SAGEConv, mean aggregation) for MI455X / gfx1250.
//
// Roofline: gather/scatter over 1.2M edges x 64ch f32 dominates (~1.2 GB of
// L2-resident traffic; whole feature matrix = 25.6 MB << 192 MB L2). The two
// dense projections per layer (N x 64 @ 64 x {64,32}) are ~0.8 GFLOP each and
// are mapped onto V_WMMA_F32_16X16X4_F32 (the f32 WMMA shape on CDNA5),
// one wave per 16x16 output tile, A tiles staged through LDS.

#define N_NODES 100000
#define N_EDGES 1200000
#define IN_CH   64
#define HID_CH  64
#define OUT_CH  32

typedef __attribute__((ext_vector_type(2))) float v2f;
typedef __attribute__((ext_vector_type(8))) float v8f;

// ---------------------------------------------------------------- zero init
__global__ void zero_f32(float* __restrict__ p, long n) {
    long i = (long)blockIdx.x * blockDim.x + threadIdx.x;
    if (i < n) p[i] = 0.0f;
}

// ------------------------------------------------------------- degree count
__global__ void count_deg(const long long* __restrict__ ei, float* __restrict__ cnt) {
    int e = blockIdx.x * blockDim.x + threadIdx.x;
    if (e < N_EDGES) {
        long long d = ei[(long)N_EDGES + e];   // edge_index[1][e] = dst
        atomicAdd(&cnt[d], 1.0f);
    }
}

// ---------------------------------------------- edge gather + scatter (64ch)
// thread t -> edge e = t/64, channel c = t%64. Lane index == channel =>
// coalesced 256B row loads from feat[src] and coalesced atomics to agg[dst].
__global__ void scatter64(const float* __restrict__ feat,
                          const long long* __restrict__ ei,
                          float* __restrict__ agg) {
    long t = (long)blockIdx.x * blockDim.x + threadIdx.x;
    if (t < (long)N_EDGES * 64) {
        int  c = (int)(t & 63);
        long e = t >> 6;
        long long s = ei[e];                    // edge_index[0][e] = src
        long long d = ei[(long)N_EDGES + e];    // edge_index[1][e] = dst
        atomicAdd(&agg[d * 64 + c], feat[s * 64 + c]);
    }
}

// -------------------------------------------------------------- WMMA GEMM
// out[16 x NCOL] tile per block = (agg/max(cnt,1)) @ Wl + self @ Wr + bias.
// Block = (NCOL/16) waves; wave w owns output columns [16w, 16w+16).
// K = 64 for both terms; V_WMMA_F32_16X16X4_F32 advances K by 4.
template <int NCOL, bool RELU>
__global__ void sage_gemm(const float* __restrict__ agg,
                          const float* __restrict__ cnt,
                          const float* __restrict__ selfF,
                          const float* __restrict__ Wl,   // [64 x NCOL] row-major
                          const float* __restrict__ Wr,   // [64 x NCOL] row-major
                          const float* __restrict__ bias, // [NCOL]
                          float* __restrict__ out) {      // [N x NCOL]
    constexpr int NW = NCOL / 16;              // waves per block
    __shared__ float sAgg[16][65];             // stride 65: bank-conflict free
    __shared__ float sX[16][65];

    const int tid  = threadIdx.x;
    const int lane = tid & 31;
    const int wave = tid >> 5;
    const int rowBase = blockIdx.x * 16;       // N_NODES == 6250*16, no tail

    // Stage A tiles: mean-scaled aggregate and root features, 16x64 each.
    for (int i = tid; i < 16 * 64; i += NW * 32) {
        int r = i >> 6, c = i & 63;
        float inv = 1.0f / fmaxf(cnt[rowBase + r], 1.0f);
        sAgg[r][c] = agg[(long)(rowBase + r) * 64 + c] * inv;
        sX[r][c]   = selfF[(long)(rowBase + r) * 64 + c];
    }
    __syncthreads();

    // A 16x4 f32 fragment layout (ISA 7.12.2): lanes 0-15 hold K=k0,k0+1 for
    // row M=lane; lanes 16-31 hold K=k0+2,k0+3 for row M=lane-16.
    // B 4x16 mirrors the C/D half-split: VGPR0 = rows K=k0 / k0+2.
    const int m     = lane & 15;
    const int khalf = (lane >> 4) * 2;         // 0 or 2
    const int ncol  = wave * 16 + (lane & 15); // output column for B/C/D

    v8f acc = {};
    #pragma unroll
    for (int k0 = 0; k0 < 64; k0 += 4) {       // aggregate term @ Wl
        v2f a, b;
        a.x = sAgg[m][k0 + khalf];
        a.y = sAgg[m][k0 + khalf + 1];
        b.x = Wl[(k0 + khalf) * NCOL + ncol];
        b.y = Wl[(k0 + khalf + 1) * NCOL + ncol];
        acc = __builtin_amdgcn_wmma_f32_16x16x4_f32(
            false, a, false, b, (short)0, acc, false, false);
    }
    #pragma unroll
    for (int k0 = 0; k0 < 64; k0 += 4) {       // root term @ Wr
        v2f a, b;
        a.x = sX[m][k0 + khalf];
        a.y = sX[m][k0 + khalf + 1];
        b.x = Wr[(k0 + khalf) * NCOL + ncol];
        b.y = Wr[(k0 + khalf + 1) * NCOL + ncol];
        acc = __builtin_amdgcn_wmma_f32_16x16x4_f32(
            false, a, false, b, (short)0, acc, false, false);
    }

    // C/D layout: VGPR j -> (M=j, N=lane) for lanes 0-15, (M=j+8, N=lane-16)
    // for lanes 16-31. Fuse bias (+ ReLU for layer 1) on writeback.
    const int mAdd = (lane >> 4) * 8;
    const float bv = bias[ncol];
    #pragma unroll
    for (int j = 0; j < 8; ++j) {
        int row = rowBase + j + mAdd;
        float v = acc[j] + bv;
        if (RELU) v = fmaxf(v, 0.0f);
        out[(long)row * NCOL + ncol] = v;
    }
}

// ---------------------------------------------------------------- launcher
extern "C" void kernel_launch(void* const* d_in, const int* in_sizes, int n_in,
                              void* d_out, int out_size, void* d_ws, size_t ws_size,
                              hipStream_t stream) {
    const float*     x    = (const float*)d_in[0];
    const long long* ei   = (const long long*)d_in[1];  // int64 edge_index [2 x E]
    const float*     W1_l = (const float*)d_in[2];
    const float*     W1_r = (const float*)d_in[3];
    const float*     b1   = (const float*)d_in[4];
    const float*     W2_l = (const float*)d_in[5];
    const float*     W2_r = (const float*)d_in[6];
    const float*     b2   = (const float*)d_in[7];
    float*           out  = (float*)d_out;

    // Workspace layout (floats): cnt[N] | agg1[N*64] | agg2[N*64] | h1[N*64]
    float* cnt  = (float*)d_ws;
    float* agg1 = cnt + N_NODES;
    float* agg2 = agg1 + (long)N_NODES * 64;
    float* h1   = agg2 + (long)N_NODES * 64;

    const long nZero = (long)N_NODES * (1 + 64 + 64);  // cnt+agg1+agg2 contiguous
    zero_f32<<<(int)((nZero + 255) / 256), 256, 0, stream>>>(cnt, nZero);

    count_deg<<<(N_EDGES + 255) / 256, 256, 0, stream>>>(ei, cnt);

    const long nScat = (long)N_EDGES * 64;
    const int  scatBlocks = (int)((nScat + 255) / 256);

    // Layer 1: aggregate x, project with WMMA, ReLU.
    scatter64<<<scatBlocks, 256, 0, stream>>>(x, ei, agg1);
    sage_gemm<HID_CH, true><<<N_NODES / 16, (HID_CH / 16) * 32, 0, stream>>>(
        agg1, cnt, x, W1_l, W1_r, b1, h1);

    // Layer 2: aggregate h1, project with WMMA (no activation).
    scatter64<<<scatBlocks, 256, 0, stream>>>(h1, ei, agg2);
    sage_gemm<OUT_CH, false><<<N_NODES / 16, (OUT_CH / 16) * 32, 0, stream>>>(
        agg2, cnt, h1, W2_l, W2_r, b2, out);
}